// GroupFullyConnectedPooling_31765578121779
// MI455X (gfx1250) — compile-verified
//
#include <hip/hip_runtime.h>
#include <hip/hip_bf16.h>

typedef __attribute__((ext_vector_type(2))) float v2f;
typedef __attribute__((ext_vector_type(8))) float v8f;
typedef __attribute__((ext_vector_type(4))) int   v4i;

#define BATCH     512
#define GROUPS    100
#define DEMB      768     // K
#define FOUT      100     // N per group
#define NCLASSES  10000

#define BM        128                 // M rows per workgroup
#define KC        24                  // K chunk staged in LDS
#define NCHUNK    (DEMB / KC)         // 32
#define ASTR      (KC + 4)            // 28 floats = 112B row, 16B aligned
#define NP        112                 // N padded to 7 tiles of 16
#define NTILES    7
#define BBUF      (NP * ASTR)         // floats per B buffer
#define ABUF      (BM * ASTR)         // floats per A buffer

#define A_F4_PER_CHUNK   (BM * KC / 4)        // 768 = 3 per thread, exact
#define B_F4_PER_CHUNK   (KC * (FOUT / 4))    // 600 -> up to 3 per thread
#define ASYNC_PER_CHUNK  3                    // A async b128 per thread per chunk

#define GLOBAL_AS __attribute__((address_space(1)))
#define LDS_AS    __attribute__((address_space(3)))

#if defined(__has_builtin)
#if __has_builtin(__builtin_amdgcn_global_load_async_to_lds_b128) && \
    __has_builtin(__builtin_amdgcn_s_wait_asynccnt)
#define USE_ASYNC 1
#else
#define USE_ASYNC 0
#endif
#else
#define USE_ASYNC 0
#endif

#if USE_ASYNC
#define WAIT_ASYNC(n) __builtin_amdgcn_s_wait_asynccnt(n)
#else
#define WAIT_ASYNC(n)
#endif

__device__ __forceinline__ void copy16_async(const float* g, float* l) {
#if USE_ASYNC
  __builtin_amdgcn_global_load_async_to_lds_b128(
      (GLOBAL_AS v4i*)(GLOBAL_AS void*)const_cast<float*>(g),
      (LDS_AS v4i*)(LDS_AS void*)l,
      /*offset=*/0, /*cpol=*/0);
#else
  *reinterpret_cast<float4*>(l) = *reinterpret_cast<const float4*>(g);
#endif
}

__global__ __launch_bounds__(256)
void group_fc_wmma_f32(const float* __restrict__ h,
                       const float* __restrict__ w,
                       const float* __restrict__ bias,
                       float* __restrict__ out) {
  __shared__ float As[2][BM][ASTR];    // As[buf][m][k]   (async-staged)
  __shared__ float Bs[2][NP][ASTR];    // Bs[buf][n][k]   (transposed, VGPR-staged)

  const int mBlock = blockIdx.x;            // 0..3
  const int g      = blockIdx.y;            // 0..99
  const int tid    = threadIdx.x;
  const int wave   = tid >> 5;              // 0..7
  const int lane   = tid & 31;
  const int lhalf  = lane >> 4;             // 0/1
  const int l16    = lane & 15;
  const int mBase  = mBlock * BM;

  const float* __restrict__ hA = h + (size_t)g * DEMB;          // + b*GROUPS*DEMB + d
  const float* __restrict__ wB = w + (size_t)g * DEMB * FOUT;   // + d*FOUT + n

  // Zero all of Bs once: rows n in [100,112) are never re-written, so they
  // stay zero for every chunk in both buffers.
  {
    float* bz = &Bs[0][0][0];
    for (int i = tid; i < 2 * BBUF; i += 256) bz[i] = 0.f;
  }

  // ---- chunk-invariant staging descriptors ----
  // A: 3 async b128 per thread, exact cover of 128x24 floats.
  const float* aSrc[3];
  float*       aDst[3];
#pragma unroll
  for (int i = 0; i < 3; ++i) {
    int idx = i * 256 + tid;                // 0..767
    int m   = idx / (KC / 4);               // /6 -> 0..127
    int kq  = idx % (KC / 4);               // 0..5
    aSrc[i] = hA + (size_t)(mBase + m) * (GROUPS * DEMB) + kq * 4;
    aDst[i] = &As[0][m][kq * 4];
  }
  // B: up to 3 float4 global loads per thread, scattered transposed into LDS.
  const float* bSrc[3];
  float*       bDst[3];                     // &Bs[0][n][k]
  bool         bVal[3];
#pragma unroll
  for (int i = 0; i < 3; ++i) {
    int idx = i * 256 + tid;                // 0..767; 600 real
    bVal[i] = idx < B_F4_PER_CHUNK;
    int cid = bVal[i] ? idx : 0;
    int k   = cid / (FOUT / 4);             // 0..23
    int nq  = cid % (FOUT / 4);             // 0..24
    bSrc[i] = wB + (size_t)k * FOUT + nq * 4;
    bDst[i] = &Bs[0][nq * 4][k];
  }

  v8f acc[NTILES];
#pragma unroll
  for (int t = 0; t < NTILES; ++t) {
    v8f z = {0.f, 0.f, 0.f, 0.f, 0.f, 0.f, 0.f, 0.f};
    acc[t] = z;
  }

  __syncthreads();   // zero-fill visible before any B scatter stores

  // ---- prologue: stage chunk 0 into buffer 0 ----
#pragma unroll
  for (int i = 0; i < 3; ++i) copy16_async(aSrc[i], aDst[i]);
#pragma unroll
  for (int i = 0; i < 3; ++i) {
    if (bVal[i]) {
      float4 v = *reinterpret_cast<const float4*>(bSrc[i]);
      bDst[i][0 * ASTR] = v.x;
      bDst[i][1 * ASTR] = v.y;
      bDst[i][2 * ASTR] = v.z;
      bDst[i][3 * ASTR] = v.w;
    }
  }

  for (int c = 0; c < NCHUNK; ++c) {
    const int buf = c & 1;
    float4 breg[3];
    if (c + 1 < NCHUNK) {
      const int nb = (c + 1) & 1;
      const int kB = (c + 1) * KC;
      // A chunk c+1: async, straight into LDS (WAR closed by barrier (2) of c-1)
#pragma unroll
      for (int i = 0; i < 3; ++i)
        copy16_async(aSrc[i] + kB, aDst[i] + nb * ABUF);
      // B chunk c+1: global -> VGPR now; LDS scatter after compute(c)
#pragma unroll
      for (int i = 0; i < 3; ++i)
        if (bVal[i])
          breg[i] = *reinterpret_cast<const float4*>(bSrc[i] + (size_t)kB * FOUT);
      WAIT_ASYNC(ASYNC_PER_CHUNK);   // A chunk c retired; c+1 still in flight
    } else {
      WAIT_ASYNC(0);
    }
    __syncthreads();                 // (1) chunk c fully in LDS for all waves

    const float* Ab = &As[0][0][0] + buf * ABUF;
    const float* Bb = &Bs[0][0][0] + buf * BBUF;
#pragma unroll
    for (int kk = 0; kk < KC; kk += 4) {
      const int kf = kk + 2 * lhalf;        // lanes 0-15: K={kk,kk+1}; 16-31: {kk+2,kk+3}
      v2f a = *reinterpret_cast<const v2f*>(Ab + (wave * 16 + l16) * ASTR + kf);
#pragma unroll
      for (int t = 0; t < NTILES; ++t) {
        v2f bf = *reinterpret_cast<const v2f*>(Bb + (t * 16 + l16) * ASTR + kf);
        acc[t] = __builtin_amdgcn_wmma_f32_16x16x4_f32(
            /*neg_a=*/false, a, /*neg_b=*/false, bf,
            /*c_mod=*/(short)0, acc[t], /*reuse_a=*/false, /*reuse_b=*/false);
      }
    }

    if (c + 1 < NCHUNK) {
      const int nb = (c + 1) & 1;
      // safe after barrier (1): nobody reads Bs[nb] until barrier (1) of c+1
#pragma unroll
      for (int i = 0; i < 3; ++i) {
        if (bVal[i]) {
          float* d = bDst[i] + nb * BBUF;
          d[0 * ASTR] = breg[i].x;
          d[1 * ASTR] = breg[i].y;
          d[2 * ASTR] = breg[i].z;
          d[3 * ASTR] = breg[i].w;
        }
      }
    }
    __syncthreads();                 // (2) compute(c) done + B(c+1) stores published
  }

  // ---- epilogue: D[m][n] VGPR r -> M = r + 8*lhalf, N = l16 ----
#pragma unroll
  for (int t = 0; t < NTILES; ++t) {
    int nl = t * 16 + l16;                  // local N, 0..111
    if (nl < FOUT) {
      int col = g * FOUT + nl;              // 0..9999
      float bv = bias[col];
#pragma unroll
      for (int r = 0; r < 8; ++r) {
        int row = mBase + wave * 16 + r + 8 * lhalf;
        out[(size_t)row * NCLASSES + col] = acc[t][r] + bv;
      }
    }
  }
}

extern "C" void kernel_launch(void* const* d_in, const int* in_sizes, int n_in,
                              void* d_out, int out_size, void* d_ws, size_t ws_size,
                              hipStream_t stream) {
  (void)in_sizes; (void)n_in; (void)d_ws; (void)ws_size; (void)out_size;
  const float* h    = (const float*)d_in[0];  // [512, 100, 768]
  const float* w    = (const float*)d_in[1];  // [100, 768, 100]
  const float* bias = (const float*)d_in[2];  // [10000]
  float* out        = (float*)d_out;          // [512, 10000]

  dim3 grid(BATCH / BM, GROUPS);
  dim3 block(256);
  group_fc_wmma_f32<<<grid, block, 0, stream>>>(h, w, bias, out);
}